// EdgeAwareGAT_10668698763493
// MI455X (gfx1250) — compile-verified
//
#include <hip/hip_runtime.h>
#include <hip/hip_bf16.h>

// EdgeAwareGAT for MI455X (gfx1250, wave32, WMMA).
// Pipeline (all fp32, WMMA f32_16x16x4 for the two GEMMs):
//  K0 zero        : zero seg_max/seg_sum/S1/acc regions of workspace
//  K1 node_gemm   : h = x @ W_node                     (WMMA)
//  K2 node_scores : a_src[n,h]=h·att_s, a_dst[n,h]=h·att_d
//  K3a watt       : w_att[32,4] = per-head W_edge @ att_e fold
//  K3b edge_base  : elogit[e,h] = edge_attr[e]·w_att[:,h]
//  K4 logit_max   : logit = a_src[src]+a_dst[dst]+elogit; lrelu; segment max
//  K5 exp_sum     : ex = exp(logit - max[dst]); segment sum
//  K6 scatter     : S1[dst]  += alpha * h[src]   (128 f32 atomics/edge)
//                   acc[dst] += alpha * edge_attr (128 f32 atomics/edge)
//  K7 finalize    : out = LN( S1 + blockdiag_head(acc @ W_edge) )  (WMMA + LN)
// Workspace requirement: ~93 MB (floats): h, a_src, a_dst, w_att, elogit,
// seg_max, seg_sum, S1, acc.

typedef __attribute__((ext_vector_type(2))) float v2f;
typedef __attribute__((ext_vector_type(8))) float v8f;

#define HEADS 4
#define DHEAD 32
#define HD    128
#define INDIM 128
#define EDIM  32
#define NEG_SLOPE 0.2f

// ---- monotone float<->uint encoding for atomic float max -------------------
__device__ __forceinline__ unsigned f2ord(float f) {
  unsigned b = __float_as_uint(f);
  return (b & 0x80000000u) ? ~b : (b | 0x80000000u);
}
__device__ __forceinline__ float ord2f(unsigned u) {
  unsigned b = (u & 0x80000000u) ? (u & 0x7fffffffu) : ~u;
  return __uint_as_float(b);
}

// ---- K0: zero fill ---------------------------------------------------------
__global__ void k_zero(float* __restrict__ p, unsigned long long n) {
  unsigned long long t = (unsigned long long)blockIdx.x * blockDim.x + threadIdx.x;
  if (t < n) p[t] = 0.0f;  // bit pattern 0 == ordered-min for seg_max too
}

// ---- K1: h = x @ W_node  (WMMA f32 16x16x4) --------------------------------
// One wave computes one 16x16 output tile; 8 waves/block cover all 128 cols.
__global__ void k_node_gemm(const float* __restrict__ x,
                            const float* __restrict__ Wn,
                            float* __restrict__ h, int nrows) {
  const int lane  = threadIdx.x & 31;
  const int wave  = threadIdx.x >> 5;
  const int m0    = blockIdx.x * 16;
  if (m0 >= nrows) return;
  const int n0    = wave * 16;
  const int l16   = lane & 15;
  const int khalf = (lane >> 4) * 2;   // lanes 16-31 hold K+2, K+3
  const int rhalf = (lane >> 4) * 8;   // C/D rows 8-15 in upper lanes

  v8f c = {};
  const float* arow = x + (size_t)(m0 + l16) * INDIM;
  for (int k = 0; k < INDIM; k += 4) {
    const int kk = k + khalf;
    v2f a; a.x = arow[kk];                  a.y = arow[kk + 1];
    v2f b; b.x = Wn[(size_t)kk * HD + n0 + l16];
           b.y = Wn[(size_t)(kk + 1) * HD + n0 + l16];
    c = __builtin_amdgcn_wmma_f32_16x16x4_f32(false, a, false, b,
                                              (short)0, c, false, false);
  }
#pragma unroll
  for (int v = 0; v < 8; ++v)
    h[(size_t)(m0 + v + rhalf) * HD + n0 + l16] = c[v];
}

// ---- K2: per-node attention scores ----------------------------------------
__global__ void k_node_scores(const float* __restrict__ h,
                              const float* __restrict__ att,
                              float* __restrict__ a_src,
                              float* __restrict__ a_dst, int n) {
  int t = blockIdx.x * blockDim.x + threadIdx.x;
  if (t >= n * HEADS) return;
  int node = t >> 2, hh = t & 3;
  const float* hr = h + (size_t)node * HD + hh * DHEAD;
  const float* as = att + hh * (3 * DHEAD);
  float s0 = 0.f, s1 = 0.f;
#pragma unroll 8
  for (int d = 0; d < DHEAD; ++d) {
    float v = hr[d];
    s0 += v * as[d];
    s1 += v * as[DHEAD + d];
  }
  a_src[t] = s0;
  a_dst[t] = s1;
}

// ---- K3a: fold W_edge into att_e -> w_att[32][4] ---------------------------
__global__ void k_watt(const float* __restrict__ We,
                       const float* __restrict__ att,
                       float* __restrict__ w_att) {
  int t = threadIdx.x;          // 128 threads: (k, h)
  int k = t >> 2, hh = t & 3;
  float s = 0.f;
#pragma unroll 8
  for (int d = 0; d < DHEAD; ++d)
    s += We[(size_t)k * HD + hh * DHEAD + d] * att[hh * (3 * DHEAD) + 2 * DHEAD + d];
  w_att[t] = s;
}

// ---- K3b: edge-attr logit contribution ------------------------------------
__global__ void k_edge_base(const float* __restrict__ ea,
                            const float* __restrict__ w_att,
                            float* __restrict__ elogit, int E) {
  int e = blockIdx.x * blockDim.x + threadIdx.x;
  if (e >= E) return;
  const float* row = ea + (size_t)e * EDIM;
  float s0 = 0.f, s1 = 0.f, s2 = 0.f, s3 = 0.f;
#pragma unroll 8
  for (int k = 0; k < EDIM; ++k) {
    float v = row[k];
    const float* w = w_att + k * 4;
    s0 += v * w[0]; s1 += v * w[1]; s2 += v * w[2]; s3 += v * w[3];
  }
  float4* o = (float4*)(elogit + (size_t)e * 4);
  *o = make_float4(s0, s1, s2, s3);
}

// ---- K4: full logit + leaky relu + segment max -----------------------------
__global__ void k_logit_max(const int* __restrict__ src,
                            const int* __restrict__ dst,
                            const float* __restrict__ a_src,
                            const float* __restrict__ a_dst,
                            float* __restrict__ elogit,
                            unsigned* __restrict__ seg_max, int E) {
  int t = blockIdx.x * blockDim.x + threadIdx.x;
  if (t >= E * HEADS) return;
  int e = t >> 2, hh = t & 3;
  int s = src[e], d = dst[e];
  float lg = a_src[s * 4 + hh] + a_dst[d * 4 + hh] + elogit[t];
  lg = lg > 0.f ? lg : NEG_SLOPE * lg;
  elogit[t] = lg;
  atomicMax(&seg_max[d * 4 + hh], f2ord(lg));
}

// ---- K5: exp + segment sum -------------------------------------------------
__global__ void k_exp_sum(const int* __restrict__ dst,
                          const unsigned* __restrict__ seg_max,
                          float* __restrict__ elogit,
                          float* __restrict__ seg_sum, int E) {
  int t = blockIdx.x * blockDim.x + threadIdx.x;
  if (t >= E * HEADS) return;
  int e = t >> 2, hh = t & 3;
  int d = dst[e];
  float ex = expf(elogit[t] - ord2f(seg_max[d * 4 + hh]));
  elogit[t] = ex;
  atomicAdd(&seg_sum[d * 4 + hh], ex);
}

// ---- K6: weighted scatter (one wave per edge) ------------------------------
__global__ void k_scatter(const int* __restrict__ src,
                          const int* __restrict__ dst,
                          const float* __restrict__ h,
                          const float* __restrict__ ea,
                          const float* __restrict__ ex,
                          const float* __restrict__ seg_sum,
                          float* __restrict__ S1,
                          float* __restrict__ acc, int E) {
  int e = blockIdx.x * 8 + (threadIdx.x >> 5);
  if (e >= E) return;
  int lane = threadIdx.x & 31;
  int s = src[e], d = dst[e];
  int f  = lane * 4;          // 4 consecutive features; all in one head
  int hh = f >> 5;
  float alpha = ex[(size_t)e * 4 + hh] / (seg_sum[d * 4 + hh] + 1e-16f);

  float4 hv = *(const float4*)(h + (size_t)s * HD + f);
  float4 ev = *(const float4*)(ea + (size_t)e * EDIM + (f & 31));
  float* S1d  = S1  + (size_t)d * HD + f;
  float* accd = acc + (size_t)d * HD + f;
  atomicAdd(S1d + 0, alpha * hv.x);
  atomicAdd(S1d + 1, alpha * hv.y);
  atomicAdd(S1d + 2, alpha * hv.z);
  atomicAdd(S1d + 3, alpha * hv.w);
  atomicAdd(accd + 0, alpha * ev.x);
  atomicAdd(accd + 1, alpha * ev.y);
  atomicAdd(accd + 2, alpha * ev.z);
  atomicAdd(accd + 3, alpha * ev.w);
}

// ---- K7: out = LN( S1 + blockdiag_head(acc @ W_edge) )  (WMMA + LN) --------
__global__ void k_finalize(const float* __restrict__ S1,
                           const float* __restrict__ acc,
                           const float* __restrict__ We,
                           const float* __restrict__ gamma,
                           const float* __restrict__ beta,
                           float* __restrict__ out, int nrows) {
  __shared__ __align__(16) float tile[16][HD + 4];
  const int lane = threadIdx.x & 31;
  const int wave = threadIdx.x >> 5;
  const int m0   = blockIdx.x * 16;
  if (m0 >= nrows) return;
  const int hh    = wave >> 1;            // head
  const int ch    = wave & 1;             // 16-col half within head
  const int l16   = lane & 15;
  const int khalf = (lane >> 4) * 2;
  const int rhalf = (lane >> 4) * 8;
  const int col   = hh * DHEAD + ch * 16 + l16;

  // C initialized with S1 (the alpha*h[src] aggregate) -> WMMA accumulates onto it
  v8f c;
#pragma unroll
  for (int v = 0; v < 8; ++v)
    c[v] = S1[(size_t)(m0 + v + rhalf) * HD + col];

  const float* arow = acc + (size_t)(m0 + l16) * HD + hh * DHEAD;
  for (int k = 0; k < DHEAD; k += 4) {
    const int kk = k + khalf;
    v2f a; a.x = arow[kk];                  a.y = arow[kk + 1];
    v2f b; b.x = We[(size_t)kk * HD + col];
           b.y = We[(size_t)(kk + 1) * HD + col];
    c = __builtin_amdgcn_wmma_f32_16x16x4_f32(false, a, false, b,
                                              (short)0, c, false, false);
  }
#pragma unroll
  for (int v = 0; v < 8; ++v)
    tile[v + rhalf][col] = c[v];
  __syncthreads();

  // LayerNorm: wave w handles rows 2w, 2w+1; each lane owns 4 features.
  const float inv128 = 1.0f / 128.0f;
#pragma unroll
  for (int rr = wave * 2; rr < wave * 2 + 2; ++rr) {
    float4 vals = *(const float4*)&tile[rr][lane * 4];
    float s = vals.x + vals.y + vals.z + vals.w;
    float q = vals.x * vals.x + vals.y * vals.y + vals.z * vals.z + vals.w * vals.w;
#pragma unroll
    for (int m = 1; m < 32; m <<= 1) {
      s += __shfl_xor(s, m, 32);
      q += __shfl_xor(q, m, 32);
    }
    float mu  = s * inv128;
    float var = q * inv128 - mu * mu;
    float inv = rsqrtf(var + 1e-5f);
    size_t o = (size_t)(m0 + rr) * HD + lane * 4;
    int g = lane * 4;
    out[o + 0] = (vals.x - mu) * inv * gamma[g + 0] + beta[g + 0];
    out[o + 1] = (vals.y - mu) * inv * gamma[g + 1] + beta[g + 1];
    out[o + 2] = (vals.z - mu) * inv * gamma[g + 2] + beta[g + 2];
    out[o + 3] = (vals.w - mu) * inv * gamma[g + 3] + beta[g + 3];
  }
}

extern "C" void kernel_launch(void* const* d_in, const int* in_sizes, int n_in,
                              void* d_out, int out_size, void* d_ws, size_t ws_size,
                              hipStream_t stream) {
  const float* x      = (const float*)d_in[0];
  const int*   ei     = (const int*)d_in[1];      // [2,E]
  const float* eattr  = (const float*)d_in[2];
  const float* Wn     = (const float*)d_in[3];
  const float* We     = (const float*)d_in[4];
  const float* att    = (const float*)d_in[5];
  const float* gamma  = (const float*)d_in[6];
  const float* beta   = (const float*)d_in[7];
  float* out = (float*)d_out;

  const int n = in_sizes[0] / INDIM;   // 50000 (multiple of 16)
  const int E = in_sizes[2] / EDIM;    // 800000
  const int* src = ei;
  const int* dst = ei + E;

  // ---- workspace layout (floats) ----
  float* ws      = (float*)d_ws;
  float* h       = ws;                             // n*128
  float* a_src   = h      + (size_t)n * HD;        // n*4
  float* a_dst   = a_src  + (size_t)n * 4;         // n*4
  float* w_att   = a_dst  + (size_t)n * 4;         // 128
  float* elogit  = w_att  + 128;                   // E*4
  float* zbase   = elogit + (size_t)E * 4;         // zeroed region starts here
  unsigned* seg_max = (unsigned*)zbase;            // n*4
  float* seg_sum = zbase  + (size_t)n * 4;         // n*4
  float* S1      = seg_sum + (size_t)n * 4;        // n*128
  float* acc     = S1     + (size_t)n * HD;        // n*128
  const unsigned long long zcount = (unsigned long long)n * (4 + 4 + HD + HD);

  const int B = 256;
  k_zero<<<(unsigned)((zcount + B - 1) / B), B, 0, stream>>>(zbase, zcount);
  k_node_gemm<<<n / 16, B, 0, stream>>>(x, Wn, h, n);
  k_node_scores<<<(n * HEADS + B - 1) / B, B, 0, stream>>>(h, att, a_src, a_dst, n);
  k_watt<<<1, 128, 0, stream>>>(We, att, w_att);
  k_edge_base<<<(E + B - 1) / B, B, 0, stream>>>(eattr, w_att, elogit, E);
  k_logit_max<<<(E * HEADS + B - 1) / B, B, 0, stream>>>(src, dst, a_src, a_dst,
                                                         elogit, seg_max, E);
  k_exp_sum<<<(E * HEADS + B - 1) / B, B, 0, stream>>>(dst, seg_max, elogit, seg_sum, E);
  k_scatter<<<(E + 7) / 8, B, 0, stream>>>(src, dst, h, eattr, elogit, seg_sum, S1, acc, E);
  k_finalize<<<n / 16, B, 0, stream>>>(S1, acc, We, gamma, beta, out, n);
}